// GCN_84834194030900
// MI455X (gfx1250) — compile-verified
//
#include <hip/hip_runtime.h>
#include <hip/hip_bf16.h>

typedef __attribute__((ext_vector_type(2))) float v2f;
typedef __attribute__((ext_vector_type(8))) float v8f;

#define TPB 256

// ---------------------------------------------------------------- utilities
__global__ void k_fill(float* __restrict__ p, float v, int n) {
    int i = blockIdx.x * TPB + threadIdx.x;
    if (i < n) p[i] = v;
}

// deg[c] += ew[e]  (deg pre-filled with 1.0 for the self loop)
__global__ void k_deg_accum(const int* __restrict__ col, const float* __restrict__ ew,
                            float* __restrict__ deg, int E) {
    int e = blockIdx.x * TPB + threadIdx.x;
    if (e < E) unsafeAtomicAdd(&deg[col[e]], ew[e]);
}

// deg -> deg^{-1/2} in place
__global__ void k_rsqrt_inplace(float* __restrict__ deg, int n) {
    int i = blockIdx.x * TPB + threadIdx.x;
    if (i < n) {
        float d = deg[i];
        deg[i] = d > 0.0f ? rsqrtf(d) : 0.0f;
    }
}

// ------------------------------------------------------- WMMA dense transforms
// xw1[N,32] = x[N,16] @ W1^T   (W1 is [32,16] row-major)
// One wave per 16-row tile; V_WMMA_F32_16X16X4_F32, K accumulated in chunks of 4.
__global__ void k_gemm_xW1(const float* __restrict__ x, const float* __restrict__ W1,
                           float* __restrict__ xw, int N) {
    const int wave = threadIdx.x >> 5;
    const int lane = threadIdx.x & 31;
    const int ntiles = (N + 15) >> 4;
    const int tile = blockIdx.x * (TPB >> 5) + wave;
    if (tile >= ntiles) return;          // whole-wave uniform: EXEC stays all-ones
    const int r0   = tile << 4;
    const int m    = lane & 15;
    const int half = lane >> 4;          // 0: K=k0..k0+1, 1: K=k0+2..k0+3
    const bool full = (r0 + 16 <= N);    // wave-uniform

    int arow = r0 + m; if (arow >= N) arow = N - 1;   // clamp (stores guarded on tail)

    // A fragments: lane m holds row r0+m; VGPR0=K even base, VGPR1=+1
    v2f a[4];
#pragma unroll
    for (int kc = 0; kc < 4; ++kc) {
        const float* ap = x + (size_t)arow * 16 + kc * 4 + 2 * half;
        a[kc].x = ap[0]; a[kc].y = ap[1];
    }

#pragma unroll
    for (int nt = 0; nt < 2; ++nt) {     // two 16-wide output tiles (F_HID=32)
        v8f c = {};
#pragma unroll
        for (int kc = 0; kc < 4; ++kc) {
            // B[k][n] = W1[n][k]; n = nt*16 + m, k = kc*4 + 2*half (+1)
            const float* bp = W1 + (size_t)(nt * 16 + m) * 16 + kc * 4 + 2 * half;
            v2f b; b.x = bp[0]; b.y = bp[1];
            c = __builtin_amdgcn_wmma_f32_16x16x4_f32(
                    false, a[kc], false, b, (short)0, c, false, false);
        }
        // D: VGPR v -> row r0 + v + 8*half, col nt*16 + m
        float* base = xw + (size_t)(r0 + 8 * half) * 32 + nt * 16 + m;
        if (full) {
#pragma unroll
            for (int v = 0; v < 8; ++v) base[(size_t)v * 32] = c[v];   // imm offsets
        } else {
#pragma unroll
            for (int v = 0; v < 8; ++v)
                if (r0 + v + 8 * half < N) base[(size_t)v * 32] = c[v];
        }
    }
}

// xw2[N,16] = h[N,32] @ W2^T   (W2 is [16,32] row-major)
__global__ void k_gemm_hW2(const float* __restrict__ h, const float* __restrict__ W2,
                           float* __restrict__ xw, int N) {
    const int wave = threadIdx.x >> 5;
    const int lane = threadIdx.x & 31;
    const int ntiles = (N + 15) >> 4;
    const int tile = blockIdx.x * (TPB >> 5) + wave;
    if (tile >= ntiles) return;
    const int r0   = tile << 4;
    const int m    = lane & 15;
    const int half = lane >> 4;
    const bool full = (r0 + 16 <= N);

    int arow = r0 + m; if (arow >= N) arow = N - 1;

    v8f c = {};
#pragma unroll
    for (int kc = 0; kc < 8; ++kc) {     // K=32 in chunks of 4
        const float* ap = h + (size_t)arow * 32 + kc * 4 + 2 * half;
        v2f a; a.x = ap[0]; a.y = ap[1];
        const float* bp = W2 + (size_t)m * 32 + kc * 4 + 2 * half;  // B[k][n]=W2[n][k]
        v2f b; b.x = bp[0]; b.y = bp[1];
        c = __builtin_amdgcn_wmma_f32_16x16x4_f32(
                false, a, false, b, (short)0, c, false, false);
    }
    float* base = xw + (size_t)(r0 + 8 * half) * 16 + m;
    if (full) {
#pragma unroll
        for (int v = 0; v < 8; ++v) base[(size_t)v * 16] = c[v];
    } else {
#pragma unroll
        for (int v = 0; v < 8; ++v)
            if (r0 + v + 8 * half < N) base[(size_t)v * 16] = c[v];
    }
}

// ------------------------------------------------------------- edge scatter
// acc[col[e], :] += xw[row[e], :] * (dis[row]*ew[e]*dis[col]); TPE threads/edge,
// each thread owns 8 contiguous features: 2x b128 gather + 8x global_atomic_add_f32.
template <int F, int TPE>
__global__ void k_scatter(const int* __restrict__ row, const int* __restrict__ col,
                          const float* __restrict__ ew, const float* __restrict__ dis,
                          const float* __restrict__ xw, float* __restrict__ acc, int E) {
    long long gid = (long long)blockIdx.x * TPB + threadIdx.x;
    int e = (int)(gid / TPE);
    if (e >= E) return;
    const int fo = (int)(gid % TPE) * (F / TPE);   // multiple of 8 floats (32B)
    const int r = row[e];
    const int c = col[e];
    const float w = dis[r] * ew[e] * dis[c];
    const float4* __restrict__ src = (const float4*)(xw + (size_t)r * F + fo);
    float* __restrict__ dst = acc + (size_t)c * F + fo;
    float4 s0 = src[0];
    float4 s1 = src[1];
    unsafeAtomicAdd(&dst[0], s0.x * w);
    unsafeAtomicAdd(&dst[1], s0.y * w);
    unsafeAtomicAdd(&dst[2], s0.z * w);
    unsafeAtomicAdd(&dst[3], s0.w * w);
    unsafeAtomicAdd(&dst[4], s1.x * w);
    unsafeAtomicAdd(&dst[5], s1.y * w);
    unsafeAtomicAdd(&dst[6], s1.z * w);
    unsafeAtomicAdd(&dst[7], s1.w * w);
}

// acc += self-loop (norm = dis^2, ew=1) + bias, optional ReLU; result in place.
template <int F, bool RELU>
__global__ void k_finish(const float* __restrict__ xw, const float* __restrict__ dis,
                         const float* __restrict__ bias, float* __restrict__ acc, int N) {
    int i = blockIdx.x * TPB + threadIdx.x;
    if (i >= N * F) return;
    const int n = i / F, f = i % F;
    const float s = dis[n];
    float v = acc[i] + xw[i] * s * s + bias[f];
    if (RELU) v = fmaxf(v, 0.0f);
    acc[i] = v;
}

// ----------------------------------------------------------------- launcher
extern "C" void kernel_launch(void* const* d_in, const int* in_sizes, int n_in,
                              void* d_out, int out_size, void* d_ws, size_t ws_size,
                              hipStream_t stream) {
    const float* x   = (const float*)d_in[0];
    const int*   ei  = (const int*)d_in[1];     // [2, E]
    const float* ew  = (const float*)d_in[2];
    const float* W1  = (const float*)d_in[3];   // [32,16]
    const float* b1  = (const float*)d_in[4];
    const float* W2  = (const float*)d_in[5];   // [16,32]
    const float* b2  = (const float*)d_in[6];

    const int N = in_sizes[0] / 16;             // x is [N,16]
    const int E = in_sizes[2];                  // edge_w is [E]
    const int* row = ei;
    const int* col = ei + E;

    // workspace layout (floats): dis[N] | xw1[N*32] | acc1/h[N*32] | xw2[N*16]
    float* dis  = (float*)d_ws;
    float* xw1  = dis  + (size_t)N;
    float* acc1 = xw1  + (size_t)N * 32;
    float* xw2  = acc1 + (size_t)N * 32;
    float* out  = (float*)d_out;                // acc2 accumulates directly here

    const int ntiles = (N + 15) / 16;
    const unsigned gemm_blocks = (unsigned)((ntiles + (TPB / 32) - 1) / (TPB / 32));
    auto grid1d = [](long long n) { return (unsigned)((n + TPB - 1) / TPB); };

    // --- normalization: deg = 1 + segment_sum(ew, col); dis = deg^{-1/2}
    k_fill<<<grid1d(N), TPB, 0, stream>>>(dis, 1.0f, N);
    k_deg_accum<<<grid1d(E), TPB, 0, stream>>>(col, ew, dis, E);
    k_rsqrt_inplace<<<grid1d(N), TPB, 0, stream>>>(dis, N);

    // --- layer 1: WMMA transform, scatter, self-loop + bias + ReLU
    k_gemm_xW1<<<gemm_blocks, TPB, 0, stream>>>(x, W1, xw1, N);
    k_fill<<<grid1d((long long)N * 32), TPB, 0, stream>>>(acc1, 0.0f, N * 32);
    k_scatter<32, 4><<<grid1d((long long)E * 4), TPB, 0, stream>>>(row, col, ew, dis, xw1, acc1, E);
    k_finish<32, true><<<grid1d((long long)N * 32), TPB, 0, stream>>>(xw1, dis, b1, acc1, N);

    // --- layer 2: WMMA transform, scatter into d_out, self-loop + bias
    k_gemm_hW2<<<gemm_blocks, TPB, 0, stream>>>(acc1, W2, xw2, N);
    k_fill<<<grid1d((long long)N * 16), TPB, 0, stream>>>(out, 0.0f, N * 16);
    k_scatter<16, 2><<<grid1d((long long)E * 2), TPB, 0, stream>>>(row, col, ew, dis, xw2, out, E);
    k_finish<16, false><<<grid1d((long long)N * 16), TPB, 0, stream>>>(xw2, dis, b2, out, N);
}